// Net_ContextualGate2_84954453115094
// MI455X (gfx1250) — compile-verified
//
#include <hip/hip_runtime.h>

// ---- problem constants (from reference) ----
#define Nn   50000
#define Ee   1600000
#define Bb   1024
#define DIN  128
#define HH1  100
#define DG   20
#define DD3  200
#define MH1  128
#define MH2  32
#define FDIM 4221      // (DG+1)*(DD3+1)
#define EPSf 1e-5f

typedef __attribute__((ext_vector_type(2))) float v2f;
typedef __attribute__((ext_vector_type(8))) float v8f;

static __device__ inline v8f wmma4(v2f a, v2f b, v8f c) {
  // V_WMMA_F32_16X16X4_F32 : D = A(16x4) * B(4x16) + C(16x16), fp32 exact
  return __builtin_amdgcn_wmma_f32_16x16x4_f32(false, a, false, b, (short)0, c,
                                               false, false);
}
static __device__ inline v8f vzero8() {
  v8f z = {0.f,0.f,0.f,0.f,0.f,0.f,0.f,0.f};
  return z;
}
// branchless guarded load: unconditional load at clamped address + select
static __device__ inline float ldsel(const float* p, int idx, int limit, int stride_k,
                                     int k) {
  int cc = (idx < limit) ? idx : (limit - 1);
  float v = p[(size_t)k * limit ? 0 : 0];  // (unused; kept simple below)
  (void)v; (void)stride_k;
  return (idx < limit) ? p[cc] : 0.f;
}

// ---------------- GEMM 1: P = feat @ W1   [N,128]x[128,100] ----------------
// one wave per 16-row tile, 7 col tiles held in registers (A loaded once)
__global__ void k_gemm_feat_w1(const float* __restrict__ A,
                               const float* __restrict__ W,
                               float* __restrict__ P) {
  int w = blockIdx.x * 4 + (threadIdx.x >> 5);
  if (w >= Nn / 16) return;
  int lane = threadIdx.x & 31;
  int hh = lane >> 4, lr = lane & 15;
  const float* arow = A + (size_t)(w * 16 + lr) * DIN;
  v8f acc[7];
#pragma unroll
  for (int t = 0; t < 7; ++t) acc[t] = vzero8();
  for (int k = 0; k < DIN; k += 4) {
    int ka = k + 2 * hh;
    v2f a; a.x = arow[ka]; a.y = arow[ka + 1];
#pragma unroll
    for (int t = 0; t < 7; ++t) {
      int c = t * 16 + lr;
      int cc = (c < HH1) ? c : (HH1 - 1);      // clamp -> address always valid
      float bx = W[ka * HH1 + cc];
      float by = W[(ka + 1) * HH1 + cc];
      v2f b;
      b.x = (c < HH1) ? bx : 0.f;              // v_cndmask, no branch
      b.y = (c < HH1) ? by : 0.f;
      acc[t] = wmma4(a, b, acc[t]);
    }
  }
  int rb = w * 16 + 8 * hh;
#pragma unroll
  for (int t = 0; t < 7; ++t) {
    int c = t * 16 + lr;
    if (c < HH1) {
#pragma unroll
      for (int i = 0; i < 8; ++i) P[(size_t)(rb + i) * HH1 + c] = acc[t][i];
    }
  }
}

// ------------- edge scatter 1: msum1[dst] += P[src], deg[dst]++ -------------
// wave per edge (grid-stride); lanes 0..24 move float4 chunks (100 floats)
__global__ void k_scatter1(const float* __restrict__ P,
                           const int* __restrict__ src,
                           const int* __restrict__ dst,
                           float* __restrict__ msum, float* __restrict__ deg) {
  int w  = blockIdx.x * (blockDim.x >> 5) + (threadIdx.x >> 5);
  int nw = gridDim.x * (blockDim.x >> 5);
  int lane = threadIdx.x & 31;
  for (int e = w; e < Ee; e += nw) {
    int s = src[e], d = dst[e];
    if (lane == 0) atomicAdd(deg + d, 1.0f);
    if (lane < 25) {
      const float4 p = *reinterpret_cast<const float4*>(P + (size_t)s * HH1 + lane * 4);
      float* o = msum + (size_t)d * HH1 + lane * 4;
      atomicAdd(o + 0, p.x); atomicAdd(o + 1, p.y);
      atomicAdd(o + 2, p.z); atomicAdd(o + 3, p.w);
    }
  }
}

// ---------------- h1 = relu(msum1/deg + b1) (written over P) ----------------
__global__ void k_h1(const float* __restrict__ msum, const float* __restrict__ deg,
                     const float* __restrict__ b1, float* __restrict__ h1) {
  int i = blockIdx.x * blockDim.x + threadIdx.x;
  if (i >= Nn * HH1) return;
  int n = i / HH1, c = i - n * HH1;
  float v = msum[i] / fmaxf(deg[n], 1.f) + b1[c];
  h1[i] = fmaxf(v, 0.f);
}

// ---------------- GEMM 2: Q = h1 @ W2  [N,100]x[100,20] ----------------
__global__ void k_gemm_h1_w2(const float* __restrict__ A,
                             const float* __restrict__ W,
                             float* __restrict__ Q) {
  int w = blockIdx.x * 4 + (threadIdx.x >> 5);
  if (w >= Nn / 16) return;
  int lane = threadIdx.x & 31;
  int hh = lane >> 4, lr = lane & 15;
  const float* arow = A + (size_t)(w * 16 + lr) * HH1;
  v8f acc[2]; acc[0] = vzero8(); acc[1] = vzero8();
  for (int k = 0; k < HH1; k += 4) {
    int ka = k + 2 * hh;
    v2f a; a.x = arow[ka]; a.y = arow[ka + 1];
#pragma unroll
    for (int t = 0; t < 2; ++t) {
      int c = t * 16 + lr;
      int cc = (c < DG) ? c : (DG - 1);
      float bx = W[ka * DG + cc];
      float by = W[(ka + 1) * DG + cc];
      v2f b;
      b.x = (c < DG) ? bx : 0.f;
      b.y = (c < DG) ? by : 0.f;
      acc[t] = wmma4(a, b, acc[t]);
    }
  }
  int rb = w * 16 + 8 * hh;
#pragma unroll
  for (int t = 0; t < 2; ++t) {
    int c = t * 16 + lr;
    if (c < DG) {
#pragma unroll
      for (int i = 0; i < 8; ++i) Q[(size_t)(rb + i) * DG + c] = acc[t][i];
    }
  }
}

// ------------- edge scatter 2: msum2[dst] += Q[src] (20 floats) -------------
// flat (edge, quad) decomposition: 5 float4 chunks per edge
__global__ void k_scatter2(const float* __restrict__ Q,
                           const int* __restrict__ src,
                           const int* __restrict__ dst,
                           float* __restrict__ msum2) {
  int t = blockIdx.x * blockDim.x + threadIdx.x;
  int stride = gridDim.x * blockDim.x;
  const int total = Ee * 5;
  for (int i = t; i < total; i += stride) {
    int e = i / 5, q = i - e * 5;
    int s = src[e], d = dst[e];
    const float4 p = *reinterpret_cast<const float4*>(Q + (size_t)s * DG + q * 4);
    float* o = msum2 + (size_t)d * DG + q * 4;
    atomicAdd(o + 0, p.x); atomicAdd(o + 1, p.y);
    atomicAdd(o + 2, p.z); atomicAdd(o + 3, p.w);
  }
}

// ---- h2 = relu(msum2/deg + b2); fused per-graph readout accumulation ----
__global__ void k_h2_hg(const float* __restrict__ msum2, const float* __restrict__ deg,
                        const float* __restrict__ b2, const int* __restrict__ gid,
                        float* __restrict__ hgsum, float* __restrict__ cnt) {
  int i = blockIdx.x * blockDim.x + threadIdx.x;
  if (i >= Nn * DG) return;
  int n = i / DG, c = i - n * DG;
  float v = fmaxf(msum2[i] / fmaxf(deg[n], 1.f) + b2[c], 0.f);
  int g = gid[n];
  atomicAdd(&hgsum[g * DG + c], v);
  if (c == 0) atomicAdd(&cnt[g], 1.f);
}

__global__ void k_hg(const float* __restrict__ hgsum, const float* __restrict__ cnt,
                     float* __restrict__ hg) {
  int i = blockIdx.x * blockDim.x + threadIdx.x;
  if (i >= Bb * DG) return;
  int b = i / DG;
  hg[i] = hgsum[i] / fmaxf(cnt[b], 1.f);
}

// --- gate GEMM: z = [hg|desc3d] @ Wg + bg ; v3 = sigmoid(z)*desc3d ---
// tile grid 64 x 13 waves, K=220; concat handled by pointer select (branchless)
__global__ void k_gate(const float* __restrict__ hg, const float* __restrict__ d3,
                       const float* __restrict__ Wg, const float* __restrict__ bg,
                       float* __restrict__ v3) {
  int w = blockIdx.x * 4 + (threadIdx.x >> 5);
  int rt = w / 13, ct = w - rt * 13;
  int lane = threadIdx.x & 31, hh = lane >> 4, lr = lane & 15;
  int m0 = rt * 16, n0 = ct * 16;
  int r = m0 + lr;
  int c = n0 + lr;
  int cc = (c < DD3) ? c : (DD3 - 1);
  v8f acc = vzero8();
  for (int k = 0; k < DG + DD3; k += 4) {
    int ka = k + 2 * hh;
    // pointer-select concat: pairs never straddle the 20-boundary (even k)
    const float* ap0 = (ka < DG) ? (hg + (size_t)r * DG + ka)
                                 : (d3 + (size_t)r * DD3 + (ka - DG));
    const float* ap1 = (ka + 1 < DG) ? (hg + (size_t)r * DG + ka + 1)
                                     : (d3 + (size_t)r * DD3 + (ka + 1 - DG));
    v2f a; a.x = *ap0; a.y = *ap1;
    float bx = Wg[(size_t)ka * DD3 + cc];
    float by = Wg[(size_t)(ka + 1) * DD3 + cc];
    v2f b;
    b.x = (c < DD3) ? bx : 0.f;
    b.y = (c < DD3) ? by : 0.f;
    acc = wmma4(a, b, acc);
  }
  if (c < DD3) {
    int rb = m0 + 8 * hh;
#pragma unroll
    for (int i = 0; i < 8; ++i) {
      float zv = acc[i] + bg[c];
      float g = 1.f / (1.f + __expf(-zv));
      v3[(size_t)(rb + i) * DD3 + c] = g * d3[(size_t)(rb + i) * DD3 + c];
    }
  }
}

// --- fusion GEMM: o1 = (hg_aug ⊗ v3_aug) @ Wf1 + bf1, A generated on the fly ---
static __device__ inline float fuseA(const float* __restrict__ hg,
                                     const float* __restrict__ v3, int r, int k) {
  int kk = (k < FDIM) ? k : (FDIM - 1);       // clamp K-pad
  int i = kk / 201;
  int j = kk - i * 201;
  int ic = (i < DG)  ? i : (DG - 1);
  int jc = (j < DD3) ? j : (DD3 - 1);
  float hv = hg[(size_t)r * DG + ic];
  float vv = v3[(size_t)r * DD3 + jc];
  float ai = (i < DG)  ? hv : 1.f;            // hg_aug: last entry = 1
  float aj = (j < DD3) ? vv : 1.f;            // v3_aug: last entry = 1
  float v  = ai * aj;
  return (k < FDIM) ? v : 0.f;
}
__global__ void k_fusion_gemm(const float* __restrict__ hg, const float* __restrict__ v3,
                              const float* __restrict__ Wf1, const float* __restrict__ bf1,
                              float* __restrict__ o1) {
  int w = blockIdx.x * 4 + (threadIdx.x >> 5);   // 512 waves: 64 row x 8 col tiles
  int rt = w >> 3, ct = w & 7;
  int lane = threadIdx.x & 31, hh = lane >> 4, lr = lane & 15;
  int r = rt * 16 + lr, c = ct * 16 + lr;
  v8f acc = vzero8();
  for (int k = 0; k < 4224; k += 4) {   // K padded from 4221
    int ka = k + 2 * hh;
    v2f a; a.x = fuseA(hg, v3, r, ka); a.y = fuseA(hg, v3, r, ka + 1);
    int k0 = (ka     < FDIM) ? ka     : (FDIM - 1);
    int k1 = (ka + 1 < FDIM) ? ka + 1 : (FDIM - 1);
    float bx = Wf1[(size_t)k0 * MH1 + c];
    float by = Wf1[(size_t)k1 * MH1 + c];
    v2f b;
    b.x = (ka     < FDIM) ? bx : 0.f;
    b.y = (ka + 1 < FDIM) ? by : 0.f;
    acc = wmma4(a, b, acc);
  }
  int rb = rt * 16 + 8 * hh;
#pragma unroll
  for (int i = 0; i < 8; ++i) o1[(size_t)(rb + i) * MH1 + c] = acc[i] + bf1[c];
}

// ---------------- batch-norm stats (biased var, like torch BN) ----------------
__global__ void k_bn_stats(const float* __restrict__ x, int cols,
                           float* __restrict__ mu, float* __restrict__ rv) {
  int c = blockIdx.x;
  int tid = threadIdx.x;
  __shared__ float ss[256], sq[256];
  float s = 0.f, q = 0.f;
  for (int r = tid; r < Bb; r += 256) {
    float v = x[(size_t)r * cols + c];
    s += v; q += v * v;
  }
  ss[tid] = s; sq[tid] = q; __syncthreads();
  for (int o = 128; o > 0; o >>= 1) {
    if (tid < o) { ss[tid] += ss[tid + o]; sq[tid] += sq[tid + o]; }
    __syncthreads();
  }
  if (tid == 0) {
    float m = ss[0] * (1.f / Bb);
    float var = sq[0] * (1.f / Bb) - m * m;
    mu[c] = m;
    rv[c] = rsqrtf(var + EPSf);
  }
}

__global__ void k_bn_apply(const float* __restrict__ x, const float* __restrict__ mu,
                           const float* __restrict__ rv, const float* __restrict__ g,
                           const float* __restrict__ b, int cols, float* __restrict__ y) {
  int i = blockIdx.x * blockDim.x + threadIdx.x;
  if (i >= Bb * cols) return;
  int c = i % cols;
  float v = (x[i] - mu[c]) * rv[c] * g[c] + b[c];
  y[i] = fmaxf(v, 0.f);
}

// ---------------- GEMM: o2 = a1 @ Wf2 + bf2  [1024,128]x[128,32] ----------------
__global__ void k_gemm_a1_wf2(const float* __restrict__ A, const float* __restrict__ W,
                              const float* __restrict__ bias, float* __restrict__ o2) {
  int w = blockIdx.x * 4 + (threadIdx.x >> 5);   // 64 row x 2 col tiles = 128 waves
  int rt = w >> 1, ct = w & 1;
  int lane = threadIdx.x & 31, hh = lane >> 4, lr = lane & 15;
  int r = rt * 16 + lr, c = ct * 16 + lr;
  v8f acc = vzero8();
  for (int k = 0; k < MH1; k += 4) {
    int ka = k + 2 * hh;
    v2f a; a.x = A[(size_t)r * MH1 + ka]; a.y = A[(size_t)r * MH1 + ka + 1];
    v2f b; b.x = W[ka * MH2 + c];         b.y = W[(ka + 1) * MH2 + c];
    acc = wmma4(a, b, acc);
  }
  int rb = rt * 16 + 8 * hh;
#pragma unroll
  for (int i = 0; i < 8; ++i) o2[(size_t)(rb + i) * MH2 + c] = acc[i] + bias[c];
}

// ---------------- final: out = a2 @ Wf3 + bf3 ----------------
__global__ void k_final(const float* __restrict__ a2, const float* __restrict__ W,
                        const float* __restrict__ b, float* __restrict__ out) {
  int r = blockIdx.x * blockDim.x + threadIdx.x;
  if (r >= Bb) return;
  float s = b[0];
#pragma unroll
  for (int j = 0; j < MH2; ++j) s += a2[(size_t)r * MH2 + j] * W[j];
  out[r] = s;
}

extern "C" void kernel_launch(void* const* d_in, const int* in_sizes, int n_in,
                              void* d_out, int out_size, void* d_ws, size_t ws_size,
                              hipStream_t stream) {
  const float* feat   = (const float*)d_in[0];
  const float* desc3d = (const float*)d_in[2];
  const float* W1  = (const float*)d_in[3];
  const float* b1  = (const float*)d_in[4];
  const float* W2  = (const float*)d_in[5];
  const float* b2  = (const float*)d_in[6];
  const float* Wg  = (const float*)d_in[7];
  const float* bg  = (const float*)d_in[8];
  const float* Wf1 = (const float*)d_in[9];
  const float* bf1 = (const float*)d_in[10];
  const float* Wf2 = (const float*)d_in[11];
  const float* bf2 = (const float*)d_in[12];
  const float* Wf3 = (const float*)d_in[13];
  const float* bf3 = (const float*)d_in[14];
  const float* gamma1 = (const float*)d_in[15];
  const float* beta1  = (const float*)d_in[16];
  const float* gamma2 = (const float*)d_in[17];
  const float* beta2  = (const float*)d_in[18];
  const int* src = (const int*)d_in[19];
  const int* dst = (const int*)d_in[20];
  const int* gid = (const int*)d_in[21];
  float* out = (float*)d_out;

  // workspace layout (floats), with aliasing:
  //   P [N*100] -> reused as h1 ; msum1 [N*100] -> reused as Q [N*20] + msum2 [N*20]
  float* ws    = (float*)d_ws;
  float* P     = ws;
  float* msum1 = ws + (size_t)Nn * HH1;
  float* Q     = msum1;
  float* msum2 = msum1 + (size_t)Nn * DG;
  float* deg   = ws + 2 * (size_t)Nn * HH1;
  float* hgsum = deg + Nn;
  float* cnt   = hgsum + Bb * DG;
  float* hg    = cnt + Bb;
  float* v3    = hg + Bb * DG;
  float* o1    = v3 + (size_t)Bb * DD3;
  float* a1    = o1 + (size_t)Bb * MH1;
  float* o2    = a1 + (size_t)Bb * MH1;
  float* a2    = o2 + (size_t)Bb * MH2;
  float* st    = a2 + (size_t)Bb * MH2;  // mu1[128] rv1[128] mu2[32] rv2[32]

  hipMemsetAsync(msum1, 0, (size_t)Nn * HH1 * sizeof(float), stream);
  hipMemsetAsync(deg,   0, (size_t)(Nn + Bb * (DG + 1)) * sizeof(float), stream);

  // layer 1 (linear-first GCN): P = feat@W1, scatter, normalize+relu
  k_gemm_feat_w1<<<(Nn / 16 + 3) / 4, 128, 0, stream>>>(feat, W1, P);
  k_scatter1<<<2048, 256, 0, stream>>>(P, src, dst, msum1, deg);
  k_h1<<<(Nn * HH1 + 255) / 256, 256, 0, stream>>>(msum1, deg, b1, P);

  // layer 2: Q = h1@W2, scatter 20-dim, normalize+relu fused with readout
  k_gemm_h1_w2<<<(Nn / 16 + 3) / 4, 128, 0, stream>>>(P, W2, Q);
  hipMemsetAsync(msum2, 0, (size_t)Nn * DG * sizeof(float), stream);
  k_scatter2<<<4096, 256, 0, stream>>>(Q, src, dst, msum2);
  k_h2_hg<<<(Nn * DG + 255) / 256, 256, 0, stream>>>(msum2, deg, b2, gid, hgsum, cnt);
  k_hg<<<(Bb * DG + 255) / 256, 256, 0, stream>>>(hgsum, cnt, hg);

  // gate + fusion + MLP head
  k_gate<<<(Bb / 16) * 13 / 4, 128, 0, stream>>>(hg, desc3d, Wg, bg, v3);
  k_fusion_gemm<<<(Bb / 16) * 8 / 4, 128, 0, stream>>>(hg, v3, Wf1, bf1, o1);
  k_bn_stats<<<MH1, 256, 0, stream>>>(o1, MH1, st, st + MH1);
  k_bn_apply<<<(Bb * MH1 + 255) / 256, 256, 0, stream>>>(o1, st, st + MH1,
                                                         gamma1, beta1, MH1, a1);
  k_gemm_a1_wf2<<<(Bb / 16) * 2 / 4, 128, 0, stream>>>(a1, Wf2, bf2, o2);
  k_bn_stats<<<MH2, 256, 0, stream>>>(o2, MH2, st + 2 * MH1, st + 2 * MH1 + MH2);
  k_bn_apply<<<(Bb * MH2 + 255) / 256, 256, 0, stream>>>(o2, st + 2 * MH1,
                                                         st + 2 * MH1 + MH2,
                                                         gamma2, beta2, MH2, a2);
  k_final<<<Bb / 256, 256, 0, stream>>>(a2, Wf3, bf3, out);
}